// GCN_48524540510801
// MI455X (gfx1250) — compile-verified
//
#include <hip/hip_runtime.h>
#include <hip/hip_bf16.h>

typedef __attribute__((ext_vector_type(2))) float v2f;
typedef __attribute__((ext_vector_type(8))) float v8f;

#define IN_CH 512
#define MID_CH 64
#define OUT_CH 40

// ---------------------------------------------------------------------------
// Utility kernels
// ---------------------------------------------------------------------------
__global__ void set_const_f(float* __restrict__ p, float v, long long n) {
    long long i = (long long)blockIdx.x * blockDim.x + threadIdx.x;
    if (i < n) p[i] = v;
}

__global__ void deg_accum(const int* __restrict__ dst, float* __restrict__ deg, int E) {
    int e = blockIdx.x * blockDim.x + threadIdx.x;
    if (e < E) atomicAdd(&deg[dst[e]], 1.0f);
}

__global__ void rsqrt_inplace(float* __restrict__ p, int n) {
    int i = blockIdx.x * blockDim.x + threadIdx.x;
    if (i < n) p[i] = rsqrtf(p[i]);
}

// ---------------------------------------------------------------------------
// GEMM1: H1[N,64] = X[N,512] @ W1[512,64]   (fp32 WMMA 16x16x4)
// block = 128 threads (4 waves); wave w computes cols [16w, 16w+16) of a
// 16-row tile. N = 100000 is divisible by 16 -> no row guard needed.
// ---------------------------------------------------------------------------
__global__ void __launch_bounds__(128)
gemm1_wmma(const float* __restrict__ X, const float* __restrict__ W,
           float* __restrict__ H, int N) {
    const int lane  = threadIdx.x & 31;
    const int wave  = threadIdx.x >> 5;
    const int row0  = blockIdx.x * 16;
    const int n0    = wave * 16;
    const int m     = lane & 15;       // matrix row (A) / col (B,C) index
    const int kh    = lane >> 4;       // K half select (0 or 1)

    v8f c = {};
    const float* xrow = X + (size_t)(row0 + m) * IN_CH;

    for (int k = 0; k < IN_CH; k += 4) {
        const int kb = k + 2 * kh;     // lanes 0-15: K=k,k+1 ; lanes 16-31: K=k+2,k+3
        v2f a;
        a.x = xrow[kb];
        a.y = xrow[kb + 1];
        v2f b;
        b.x = W[(size_t)kb * MID_CH + n0 + m];
        b.y = W[(size_t)(kb + 1) * MID_CH + n0 + m];
        c = __builtin_amdgcn_wmma_f32_16x16x4_f32(
                /*neg_a=*/false, a, /*neg_b=*/false, b,
                /*c_mod=*/(short)0, c, /*reuse_a=*/false, /*reuse_b=*/false);
    }

    const int col   = n0 + m;
    const int rbase = row0 + 8 * kh;   // C layout: lanes 16-31 hold M=8..15
#pragma unroll
    for (int r = 0; r < 8; ++r)
        H[(size_t)(rbase + r) * MID_CH + col] = c[r];
}

// ---------------------------------------------------------------------------
// GEMM2: H2[N,40] = Helu[N,64] @ W2[64,40]  (cols padded to 48, masked)
// block = 32 threads (1 wave); grid (N/16, 3)
// ---------------------------------------------------------------------------
__global__ void __launch_bounds__(32)
gemm2_wmma(const float* __restrict__ Hin, const float* __restrict__ W2,
           float* __restrict__ H2, int N) {
    const int lane = threadIdx.x & 31;
    const int row0 = blockIdx.x * 16;
    const int n0   = blockIdx.y * 16;
    const int m    = lane & 15;
    const int kh   = lane >> 4;
    const int col  = n0 + m;
    const bool cok = (col < OUT_CH);

    v8f c = {};
    const float* hrow = Hin + (size_t)(row0 + m) * MID_CH;

    for (int k = 0; k < MID_CH; k += 4) {
        const int kb = k + 2 * kh;
        v2f a;
        a.x = hrow[kb];
        a.y = hrow[kb + 1];
        v2f b;
        b.x = cok ? W2[(size_t)kb * OUT_CH + col] : 0.0f;
        b.y = cok ? W2[(size_t)(kb + 1) * OUT_CH + col] : 0.0f;
        c = __builtin_amdgcn_wmma_f32_16x16x4_f32(
                false, a, false, b, (short)0, c, false, false);
    }

    if (cok) {
        const int rbase = row0 + 8 * kh;
#pragma unroll
        for (int r = 0; r < 8; ++r)
            H2[(size_t)(rbase + r) * OUT_CH + col] = c[r];
    }
}

// ---------------------------------------------------------------------------
// Edge aggregation: AGG[dst, c] += H[src, c] * dinv[src] * dinv[dst]
// One thread per (edge, channel); consecutive threads share an edge so the
// gather of H[src,*] and the atomic scatter to AGG[dst,*] are contiguous.
// H (<=25.6MB) fits in the 192MB L2, so atomics resolve on-chip.
// ---------------------------------------------------------------------------
template <int C>
__global__ void agg_edges(const float* __restrict__ H, const float* __restrict__ dinv,
                          const int* __restrict__ src, const int* __restrict__ dst,
                          float* __restrict__ AGG, long long total) {
    long long idx = (long long)blockIdx.x * blockDim.x + threadIdx.x;
    if (idx >= total) return;
    const int e  = (int)(idx / C);
    const int ch = (int)(idx % C);
    const int s  = src[e];
    const int d  = dst[e];
    const float norm = dinv[s] * dinv[d];
    atomicAdd(&AGG[(size_t)d * C + ch], H[(size_t)s * C + ch] * norm);
}

// ---------------------------------------------------------------------------
// Layer-1 epilogue: AGG <- elu(AGG + H*dinv^2 + b)   (in place)
// ---------------------------------------------------------------------------
__global__ void selfloop_bias_elu(float* __restrict__ AGG, const float* __restrict__ H,
                                  const float* __restrict__ dinv,
                                  const float* __restrict__ bias, int N) {
    int idx = blockIdx.x * blockDim.x + threadIdx.x;
    if (idx >= N * MID_CH) return;
    const int n  = idx >> 6;
    const int ch = idx & (MID_CH - 1);
    const float di = dinv[n];
    float v = AGG[idx] + H[idx] * di * di + bias[ch];
    AGG[idx] = (v > 0.0f) ? v : (__expf(v) - 1.0f);
}

// ---------------------------------------------------------------------------
// Layer-2 epilogue + log_softmax over 40 channels. One wave32 per node:
// lane holds channel `lane` and (for lanes 0-7) channel lane+32.
// ---------------------------------------------------------------------------
__global__ void __launch_bounds__(256)
final_logsoftmax(const float* __restrict__ AGG2, const float* __restrict__ H2,
                 const float* __restrict__ dinv, const float* __restrict__ b2,
                 float* __restrict__ OUT, int N) {
    const int lane = threadIdx.x & 31;
    const int node = blockIdx.x * (blockDim.x >> 5) + (threadIdx.x >> 5);
    if (node >= N) return;

    const float di = dinv[node];
    const float d2 = di * di;
    const float* a = AGG2 + (size_t)node * OUT_CH;
    const float* h = H2   + (size_t)node * OUT_CH;

    float v0 = a[lane] + h[lane] * d2 + b2[lane];
    float v1 = -1e30f;
    if (lane < OUT_CH - 32)
        v1 = a[lane + 32] + h[lane + 32] * d2 + b2[lane + 32];

    float mx = fmaxf(v0, v1);
#pragma unroll
    for (int off = 16; off; off >>= 1)
        mx = fmaxf(mx, __shfl_xor(mx, off, 32));

    float s = __expf(v0 - mx) + ((lane < OUT_CH - 32) ? __expf(v1 - mx) : 0.0f);
#pragma unroll
    for (int off = 16; off; off >>= 1)
        s += __shfl_xor(s, off, 32);

    const float ls = __logf(s);
    OUT[(size_t)node * OUT_CH + lane] = v0 - mx - ls;
    if (lane < OUT_CH - 32)
        OUT[(size_t)node * OUT_CH + lane + 32] = v1 - mx - ls;
}

// ---------------------------------------------------------------------------
// Host launcher
// ---------------------------------------------------------------------------
extern "C" void kernel_launch(void* const* d_in, const int* in_sizes, int n_in,
                              void* d_out, int out_size, void* d_ws, size_t ws_size,
                              hipStream_t stream) {
    const float* x   = (const float*)d_in[0];
    const int*   adj = (const int*)d_in[1];
    const float* W1  = (const float*)d_in[2];
    const float* b1  = (const float*)d_in[3];
    const float* W2  = (const float*)d_in[4];
    const float* b2  = (const float*)d_in[5];
    float* out = (float*)d_out;

    const int N = in_sizes[0] / IN_CH;       // 100000
    const int E = in_sizes[1] / 2;           // 1600000
    const int* src = adj;
    const int* dst = adj + E;

    // Workspace layout (floats)
    float* dinv = (float*)d_ws;                       // N
    float* h1   = dinv + N;                           // N*64 (later reused as H2: N*40)
    float* agg1 = h1 + (size_t)N * MID_CH;            // N*64 (in-place ELU -> GEMM2 input)
    float* agg2 = agg1 + (size_t)N * MID_CH;          // N*40
    float* h2   = h1;                                 // alias: h1 dead after ELU

    // 1) degree with self loops -> dinv = rsqrt(deg)
    set_const_f<<<(N + 255) / 256, 256, 0, stream>>>(dinv, 1.0f, N);
    deg_accum<<<(E + 255) / 256, 256, 0, stream>>>(dst, dinv, E);
    rsqrt_inplace<<<(N + 255) / 256, 256, 0, stream>>>(dinv, N);

    // 2) zero accumulators
    {
        long long n1 = (long long)N * MID_CH;
        long long n2 = (long long)N * OUT_CH;
        set_const_f<<<(unsigned)((n1 + 255) / 256), 256, 0, stream>>>(agg1, 0.0f, n1);
        set_const_f<<<(unsigned)((n2 + 255) / 256), 256, 0, stream>>>(agg2, 0.0f, n2);
    }

    // 3) H1 = X @ W1 via fp32 WMMA
    gemm1_wmma<<<N / 16, 128, 0, stream>>>(x, W1, h1, N);

    // 4) edge aggregation layer 1
    {
        long long total = (long long)E * MID_CH;
        agg_edges<MID_CH><<<(unsigned)((total + 255) / 256), 256, 0, stream>>>(
            h1, dinv, src, dst, agg1, total);
    }

    // 5) self-loop + bias + ELU (in place into agg1)
    selfloop_bias_elu<<<((N * MID_CH) + 255) / 256, 256, 0, stream>>>(agg1, h1, dinv, b1, N);

    // 6) H2 = Helu @ W2 via fp32 WMMA (cols padded to 48, masked to 40)
    {
        dim3 grid(N / 16, 3);
        gemm2_wmma<<<grid, 32, 0, stream>>>(agg1, W2, h2, N);
    }

    // 7) edge aggregation layer 2
    {
        long long total = (long long)E * OUT_CH;
        agg_edges<OUT_CH><<<(unsigned)((total + 255) / 256), 256, 0, stream>>>(
            h2, dinv, src, dst, agg2, total);
    }

    // 8) self-loop + bias + log_softmax -> out
    final_logsoftmax<<<(N + 7) / 8, 256, 0, stream>>>(agg2, h2, dinv, b2, out, N);
}